// HebbianMap2d_56942676410848
// MI455X (gfx1250) — compile-verified
//
#include <hip/hip_runtime.h>
#include <math.h>

// ---------------- problem constants (from reference) ----------------
constexpr int Bn = 32, Cc = 3, Hh = 64, Ww = 64;
constexpr int KH = 5, KW = 5;
constexpr int Ho = 60, Wo = 60;
constexpr int O  = 256;          // SOM units (16x16 grid)
constexpr int D  = 75;           // C*KH*KW
constexpr int Dp = 80;           // D padded to multiple of 16 (and of 4 for K-steps)
constexpr int Ntot = Bn * Ho * Wo;          // 115200 pixels
constexpr int TPIX = 16;                    // pixels per workgroup (one wave32)
constexpr float ETA = 0.01f;

typedef __attribute__((ext_vector_type(2))) float v2f;
typedef __attribute__((ext_vector_type(8))) float v8f;

// D = A(16x4) * B(4x16) + C, full f32 WMMA (CDNA5 V_WMMA_F32_16X16X4_F32)
static __device__ inline v8f wmma_f32(v2f a, v2f b, v8f c) {
    return __builtin_amdgcn_wmma_f32_16x16x4_f32(
        /*neg_a=*/false, a, /*neg_b=*/false, b,
        /*c_mod=*/(short)0, c, /*reuse_a=*/false, /*reuse_b=*/false);
}

// ---------------- workspace layout (floats) ----------------
// Wp : [O][Dp]   padded weights            offset 0        (20480 f)
// rx : [O][Dp]   r^T @ x_unf accumulator   offset 20480    (20480 f)
// rr : [O][O]    r^T @ r accumulator       offset 40960    (65536 f)
constexpr int WS_WP = 0;
constexpr int WS_RX = O * Dp;
constexpr int WS_RR = 2 * O * Dp;

// ============================================================
// Kernel 0: zero accumulators, build padded weight matrix
// ============================================================
__global__ void hebb_init(const float* __restrict__ w, float* __restrict__ ws) {
    int idx = blockIdx.x * blockDim.x + threadIdx.x;   // 0 .. 65535
    if (idx < O * Dp) {
        int o = idx / Dp, d = idx % Dp;
        ws[WS_WP + idx] = (d < D) ? w[o * D + d] : 0.0f;
        ws[WS_RX + idx] = 0.0f;
    }
    if (idx < O * O) ws[WS_RR + idx] = 0.0f;
}

// ============================================================
// Kernel 1: one wave32 workgroup per 16-pixel tile.
//  GEMM1 (WMMA f32) -> relu -> y_out, rowmax/winners/lfb -> r,
//  then WMMA rank-16 updates of rx and tril(rr) via f32 atomics.
// ============================================================
__global__ void __launch_bounds__(32)
hebb_tile(const float* __restrict__ x, const float* __restrict__ ws,
          float* __restrict__ y_out, float* __restrict__ rx, float* __restrict__ rr) {
    __shared__ float Xs[TPIX][Dp];          //  5 KB  x_unf tile (zero padded)
    __shared__ float Ys[TPIX][O];           // 16 KB  y, later overwritten with r
    __shared__ unsigned short Wl[TPIX][O];  //  8 KB  winner index lists
    __shared__ int   Wc[TPIX];
    __shared__ float Rmax[TPIX];

    const float* Wp = ws + WS_WP;
    const int lane = threadIdx.x;           // 0..31
    const int l15  = lane & 15;
    const int half = lane >> 4;             // 0/1
    const int koff = half * 2;              // per-lane K sub-offset in f32 WMMA frags
    const int tile0 = blockIdx.x * TPIX;    // first pixel index of this tile

    // ---- stage x_unf tile into LDS (unfold on the fly, channel-major c,ki,kj) ----
    for (int idx = lane; idx < TPIX * Dp; idx += 32) {
        int m = idx / Dp, d = idx % Dp;
        float v = 0.0f;
        if (d < D) {
            int n  = tile0 + m;
            int b  = n / (Ho * Wo);
            int rm = n % (Ho * Wo);
            int ho = rm / Wo, wo = rm % Wo;
            int c  = d / (KH * KW);
            int r2 = d % (KH * KW);
            int ki = r2 / KW, kj = r2 % KW;
            v = x[((b * Cc + c) * Hh + (ho + ki)) * Ww + (wo + kj)];
        }
        Xs[m][d] = v;
    }
    __syncthreads();

    // ---- GEMM1: S[16 x 256] = Xs[16 x Dp] @ Wp^T, K-steps of 4 ----
    v8f acc[16];
#pragma unroll
    for (int t = 0; t < 16; ++t) acc[t] = (v8f)(0.0f);

    for (int kk = 0; kk < Dp; kk += 4) {
        v2f a;
        a[0] = Xs[l15][kk + koff + 0];
        a[1] = Xs[l15][kk + koff + 1];
#pragma unroll
        for (int t = 0; t < 16; ++t) {
            const float* wrow = Wp + (t * 16 + l15) * Dp + kk + koff;
            v2f b; b[0] = wrow[0]; b[1] = wrow[1];
            acc[t] = wmma_f32(a, b, acc[t]);
        }
    }

    // ---- relu, spill y tile to LDS (element (m,o): m = vg+8*half, o = t*16+l15) ----
#pragma unroll
    for (int t = 0; t < 16; ++t)
#pragma unroll
        for (int vg = 0; vg < 8; ++vg) {
            float y = acc[t][vg];
            Ys[vg + 8 * half][t * 16 + l15] = y > 0.0f ? y : 0.0f;
        }
    __syncthreads();

    // ---- store y_out (NCHW: [B][O][Ho][Wo]) ----
    for (int idx = lane; idx < TPIX * O; idx += 32) {
        int o = idx >> 4, m = idx & 15;
        int n  = tile0 + m;
        int b  = n / (Ho * Wo);
        int rm = n % (Ho * Wo);
        int ho = rm / Wo, wo = rm % Wo;
        y_out[((b * O + o) * Ho + ho) * Wo + wo] = Ys[m][o];
    }

    // ---- per-pixel row max (each half of the wave scans 128 columns) ----
    {
        float mx = 0.0f;                       // y >= 0 so 0 is a safe identity
        for (int c = 0; c < 128; ++c) mx = fmaxf(mx, Ys[l15][half * 128 + c]);
        mx = fmaxf(mx, __shfl_xor(mx, 16, 32));
        if (lane < 16) { Rmax[lane] = mx; Wc[lane] = 0; }
    }
    __syncthreads();

    // ---- winner collection (exact tie handling incl. all-zero rows) ----
    for (int c = 0; c < 128; ++c) {
        int o = half * 128 + c;
        if (Ys[l15][o] == Rmax[l15]) {
            int slot = atomicAdd(&Wc[l15], 1);
            Wl[l15][slot] = (unsigned short)o;
        }
    }
    __syncthreads();

    // ---- lateral feedback table: gv[c] = exp(-c^2 / (2*sigma^2)), sigma=7 ----
    float gv[9];
#pragma unroll
    for (int c = 0; c < 9; ++c) gv[c] = expf(-(float)(c * c) / 98.0f);

    // ---- r = y * clip(sum_winners G(o, o'), -1, 1), overwrite Ys in place ----
    for (int idx = lane; idx < TPIX * O; idx += 32) {
        int o = idx >> 4, m = idx & 15;
        float y = Ys[m][o];
        float r = 0.0f;
        if (y > 0.0f) {
            int oi = o >> 4, oj = o & 15;
            float s = 0.0f;
            int cnt = Wc[m];
            for (int wI = 0; wI < cnt; ++wI) {
                int ow = Wl[m][wI];
                int di = (ow >> 4) - oi, dj = (ow & 15) - oj;
                if (di >= -7 && di <= 8 && dj >= -7 && dj <= 8) {
                    int adi = di < 0 ? -di : di;
                    int adj = dj < 0 ? -dj : dj;
                    s += gv[adi > adj ? adi : adj];
                }
            }
            r = y * fminf(s, 1.0f);            // s >= 0, lower clip is a no-op
        }
        Ys[m][o] = r;
    }
    __syncthreads();

    // ---- rx += r^T @ Xs  (A = r^T [O x 16], B = Xs [16 x Dp]) ----
    for (int ot = 0; ot < 16; ++ot) {
        v8f ar[5];
#pragma unroll
        for (int dt = 0; dt < 5; ++dt) ar[dt] = (v8f)(0.0f);
        for (int kk = 0; kk < 16; kk += 4) {
            v2f a;
            a[0] = Ys[kk + koff + 0][ot * 16 + l15];
            a[1] = Ys[kk + koff + 1][ot * 16 + l15];
#pragma unroll
            for (int dt = 0; dt < 5; ++dt) {
                v2f b;
                b[0] = Xs[kk + koff + 0][dt * 16 + l15];
                b[1] = Xs[kk + koff + 1][dt * 16 + l15];
                ar[dt] = wmma_f32(a, b, ar[dt]);
            }
        }
#pragma unroll
        for (int dt = 0; dt < 5; ++dt)
#pragma unroll
            for (int vg = 0; vg < 8; ++vg) {
                int o = ot * 16 + vg + 8 * half;
                int d = dt * 16 + l15;
                atomicAdd(&rx[o * Dp + d], ar[dt][vg]);
            }
    }

    // ---- rr += r^T @ r, lower-triangular tiles only (ti >= tj) ----
    for (int ti = 0; ti < 16; ++ti) {
        for (int tj = 0; tj <= ti; ++tj) {
            v8f ac = (v8f)(0.0f);
            for (int kk = 0; kk < 16; kk += 4) {
                v2f a, b;
                a[0] = Ys[kk + koff + 0][ti * 16 + l15];
                a[1] = Ys[kk + koff + 1][ti * 16 + l15];
                b[0] = Ys[kk + koff + 0][tj * 16 + l15];
                b[1] = Ys[kk + koff + 1][tj * 16 + l15];
                ac = wmma_f32(a, b, ac);
            }
#pragma unroll
            for (int vg = 0; vg < 8; ++vg) {
                int oa = ti * 16 + vg + 8 * half;
                int ob = tj * 16 + l15;
                atomicAdd(&rr[oa * O + ob], ac[vg]);
            }
        }
    }
}

// ============================================================
// Kernel 2: new_w[o,d] = w[o,d] + ETA * (rx[o,d] - sum_{b<=o} rr[o,b]*w[b,d]) / N
// ============================================================
__global__ void hebb_finalize(const float* __restrict__ w,
                              const float* __restrict__ rx,
                              const float* __restrict__ rr,
                              float* __restrict__ out_w) {
    int idx = blockIdx.x * blockDim.x + threadIdx.x;
    if (idx >= O * D) return;
    int o = idx / D, d = idx % D;
    float acc = rx[o * Dp + d];
    const float* rrow = rr + o * O;
    for (int b = 0; b <= o; ++b) acc -= rrow[b] * w[b * D + d];   // tril(rr) @ w
    out_w[idx] = w[idx] + ETA * acc / (float)Ntot;
}

// ============================================================
extern "C" void kernel_launch(void* const* d_in, const int* in_sizes, int n_in,
                              void* d_out, int out_size, void* d_ws, size_t ws_size,
                              hipStream_t stream) {
    (void)in_sizes; (void)n_in; (void)out_size; (void)ws_size;
    const float* x = (const float*)d_in[0];      // [32,3,64,64]
    const float* w = (const float*)d_in[1];      // [256,3,5,5]
    float* out   = (float*)d_out;
    float* y_out = out;                                    // [32,256,60,60]
    float* out_w = out + (size_t)Bn * O * Ho * Wo;         // [256,75]
    float* ws = (float*)d_ws;
    float* rx = ws + WS_RX;
    float* rr = ws + WS_RR;

    hebb_init<<<(O * O + 255) / 256, 256, 0, stream>>>(w, ws);
    hebb_tile<<<Ntot / TPIX, 32, 0, stream>>>(x, ws, y_out, rx, rr);
    hebb_finalize<<<(O * D + 255) / 256, 256, 0, stream>>>(w, rx, rr, out_w);
}